// PPNM_Layer_90787018703189
// MI455X (gfx1250) — compile-verified
//
#include <hip/hip_runtime.h>
#include <hip/hip_bf16.h>

// PPNM layer: out = mat + b_s[:,None]*mat^2, mat = x[B,16] @ W[16,224]
// B=262144, E=16, L=224. Memory-bound (~253MB traffic, ~11us floor at 23.3TB/s).
// Strategy: f32 WMMA (V_WMMA_F32_16X16X4_F32), K=16 as 4 chained K=4 steps,
// epilogue fused in accumulator registers, NT stores for the 235MB output.

#define PPNM_B 262144
#define PPNM_E 16
#define PPNM_L 224

typedef __attribute__((ext_vector_type(2))) float v2f;
typedef __attribute__((ext_vector_type(8))) float v8f;

__launch_bounds__(256, 4)
__global__ void ppnm_wmma_f32(const float* __restrict__ x,
                              const float* __restrict__ W,
                              const float* __restrict__ bsv,
                              float* __restrict__ out)
{
    const int lane   = threadIdx.x & 31;
    const int wave   = threadIdx.x >> 5;
    const int half   = lane >> 4;      // 0: lanes 0-15, 1: lanes 16-31
    const int lane16 = lane & 15;

    const int tile_n = blockIdx.x;                 // 0..13  (224/16)
    const int tile_m = blockIdx.y * 8 + wave;      // 0..16383 (262144/16)
    const int m0 = tile_m * 16;
    const int n0 = tile_n * 16;

    // A-fragment row for this lane (16x4 f32 A layout: lanes 0-15 & 16-31 both M=0..15)
    const int arow = m0 + lane16;
    // B-fragment column for this lane (4x16 f32 B: row K striped across lanes)
    const int bcol = n0 + lane16;

    const float* __restrict__ xrow = x + (size_t)arow * PPNM_E;

    v8f c = {};
    #pragma unroll
    for (int s = 0; s < 4; ++s) {
        // K-range for this step: 4s..4s+3.
        // A layout (32-bit A 16x4): VGPR0 = K=4s+2*half, VGPR1 = K=4s+2*half+1
        const int ka = 4 * s + 2 * half;
        v2f a = __builtin_nontemporal_load((const v2f*)(xrow + ka)); // 8B aligned: ka even

        // B layout (4x16 f32): VGPR0 lanes(lo)=K0 / lanes(hi)=K2, VGPR1 = K1/K3
        v2f b;
        b.x = W[(size_t)(ka + 0) * PPNM_L + bcol];
        b.y = W[(size_t)(ka + 1) * PPNM_L + bcol];

        // D = A*B + C, full f32 (matches reference f32 matmul); RNE rounding.
        c = __builtin_amdgcn_wmma_f32_16x16x4_f32(
                /*neg_a=*/false, a, /*neg_b=*/false, b,
                /*c_mod=*/(short)0, c, /*reuse_a=*/false, /*reuse_b=*/false);
    }

    // Fused PPNM epilogue on the accumulator tile.
    // C/D layout: VGPR v -> row (v + 8*half), col = n0 + lane16.
    const float* __restrict__ bsp = bsv + m0 + 8 * half;  // broadcast within half-wave
    float* __restrict__ obase = out + (size_t)(m0 + 8 * half) * PPNM_L + (n0 + lane16);

    #pragma unroll
    for (int v = 0; v < 8; ++v) {
        float m  = c[v];
        float bc = bsp[v];
        float r  = m + bc * (m * m);
        __builtin_nontemporal_store(r, obase + (size_t)v * PPNM_L);
    }
}

extern "C" void kernel_launch(void* const* d_in, const int* in_sizes, int n_in,
                              void* d_out, int out_size, void* d_ws, size_t ws_size,
                              hipStream_t stream) {
    const float* x   = (const float*)d_in[0];   // [B, E]
    const float* W   = (const float*)d_in[1];   // [E, L]
    const float* b_s = (const float*)d_in[2];   // [B]
    float* out = (float*)d_out;                 // [B, L]

    dim3 block(256, 1, 1);                      // 8 waves
    dim3 grid(PPNM_L / 16, PPNM_B / (16 * 8), 1); // (14, 2048)
    hipLaunchKernelGGL(ppnm_wmma_f32, grid, block, 0, stream, x, W, b_s, out);
}